// EMMA_23321672417406
// MI455X (gfx1250) — compile-verified
//
#include <hip/hip_runtime.h>
#include <hip/hip_bf16.h>

typedef float v4f __attribute__((ext_vector_type(4)));
typedef int   v4i __attribute__((ext_vector_type(4)));

#define MODES 8
#define E_CONSTF 2.718281828459045f

#if __has_builtin(__builtin_amdgcn_global_load_async_to_lds_b128)
#define HAVE_ASYNC_LDS 1
#else
#define HAVE_ASYNC_LDS 0
#endif

__device__ __forceinline__ void wait_asynccnt0() {
#if __has_builtin(__builtin_amdgcn_s_wait_asynccnt)
    __builtin_amdgcn_s_wait_asynccnt(0);
#else
    asm volatile("s_wait_asynccnt 0x0" ::: "memory");
#endif
}
__device__ __forceinline__ void wait_asynccnt1() {
#if __has_builtin(__builtin_amdgcn_s_wait_asynccnt)
    __builtin_amdgcn_s_wait_asynccnt(1);
#else
    asm volatile("s_wait_asynccnt 0x1" ::: "memory");
#endif
}

#if HAVE_ASYNC_LDS
__device__ __forceinline__ void async_b128_to_lds(const v4f* g, v4f* lds_generic) {
    __builtin_amdgcn_global_load_async_to_lds_b128(
        (__attribute__((address_space(1))) v4i*)g,
        (__attribute__((address_space(3))) v4i*)lds_generic,
        0, 0);
}
#endif

// ---------------------------------------------------------------------------
// Kernel 1: per-(row,mode) beta computation. 8 lanes == 1 row (wave32 -> 4
// rows per wave). Writes beta[N*M] to workspace.
// ---------------------------------------------------------------------------
__global__ void EMMA_beta_kernel(const float* __restrict__ q,
                                 const float* __restrict__ w,
                                 const float* __restrict__ b,
                                 const float* __restrict__ gammas,
                                 const float* __restrict__ G,
                                 const float* __restrict__ bias,
                                 const float* __restrict__ qmin,
                                 float* __restrict__ beta_out,
                                 int n_rows)
{
    __shared__ float s_gsym[MODES * MODES];
    __shared__ float s_w[MODES], s_b[MODES], s_G[MODES], s_bias[MODES], s_qmin[MODES];

    const int t = threadIdx.x;
    if (t < MODES * MODES) {
        int i = t >> 3, j = t & 7;
        // symmetrized gamma: g[i,j] = gammas[min(i,j), max(i,j)]
        s_gsym[t] = (i < j) ? gammas[i * MODES + j] : gammas[j * MODES + i];
    }
    if (t < MODES) {
        s_w[t] = w[t]; s_b[t] = b[t]; s_G[t] = G[t];
        s_bias[t] = bias[t]; s_qmin[t] = qmin[t];
    }
    __syncthreads();

    const long long gid = (long long)blockIdx.x * blockDim.x + t;  // (row, mode)
    const int m = (int)(gid & (MODES - 1));
    const long long row = gid >> 3;
    if (row >= n_rows) return;

    const float e = E_CONSTF;
    float corr = fmaxf(q[gid] - s_qmin[m] + e, e);
    float phi  = fmaf(1.0f + s_w[m], corr, s_b[m]);      // > 0 guaranteed
    float L    = __logf(phi);

    // modal_i = sum_j exp(L_j + g[i][j]*(L_i - L_j))
    float modal = 0.0f;
    #pragma unroll
    for (int j = 0; j < MODES; ++j) {
        float Lj = __shfl(L, j, MODES);                  // width-8 subgroup
        float g  = s_gsym[m * MODES + j];
        modal += __expf(fmaf(g, L - Lj, Lj));
    }

    // softmax over the 8-lane group of (-modal), stabilized by min(modal)
    float mn = modal;
    #pragma unroll
    for (int s = 1; s < MODES; s <<= 1) mn = fminf(mn, __shfl_xor(mn, s, MODES));
    float num = __expf(mn - modal);
    float den = num;
    #pragma unroll
    for (int s = 1; s < MODES; s <<= 1) den += __shfl_xor(den, s, MODES);
    float alpha = num / den;

    float beta = tanhf(fmaf(s_G[m], alpha, s_bias[m]));
    beta_out[gid] = fmaxf(beta, 0.0f);
}

// ---------------------------------------------------------------------------
// Kernel 2: out[n,m,d] = beta[n,m] * x[n,m,d]. Streaming, double-buffered
// async global->LDS DMA (each lane reads back only its own LDS slot, so the
// only required sync is s_wait_asynccnt — no barriers). NT stores: data is
// touched exactly once and exceeds the 192 MB L2.
// ---------------------------------------------------------------------------
__global__ void EMMA_scale_kernel(const float* __restrict__ x,
                                  const float* __restrict__ beta,
                                  float* __restrict__ out,
                                  long long total4)
{
    const v4f* __restrict__ x4 = (const v4f*)x;
    v4f* __restrict__ o4 = (v4f*)out;

    const long long stride = (long long)gridDim.x * blockDim.x;
    long long i = (long long)blockIdx.x * blockDim.x + threadIdx.x;
    const int t = threadIdx.x;

#if HAVE_ASYNC_LDS
    __shared__ v4f stage[2][256];
    int buf = 0;
    if (i < total4) {
        async_b128_to_lds(x4 + i, &stage[buf][t]);
    }
    for (; i < total4; i += stride) {
        const long long nxt = i + stride;
        if (nxt < total4) {
            async_b128_to_lds(x4 + nxt, &stage[buf ^ 1][t]);
            wait_asynccnt1();          // in-order: current tile has landed
        } else {
            wait_asynccnt0();
        }
        v4f v = stage[buf][t];
        float bt = beta[i >> 5];       // D/4 = 32 float4 per (n,m): wave-uniform
        v *= bt;
        __builtin_nontemporal_store(v, o4 + i);
        buf ^= 1;
    }
#else
    for (; i < total4; i += stride) {
        v4f v = __builtin_nontemporal_load(x4 + i);
        float bt = beta[i >> 5];
        v *= bt;
        __builtin_nontemporal_store(v, o4 + i);
    }
#endif
}

// ---------------------------------------------------------------------------
extern "C" void kernel_launch(void* const* d_in, const int* in_sizes, int n_in,
                              void* d_out, int out_size, void* d_ws, size_t ws_size,
                              hipStream_t stream) {
    const float* x      = (const float*)d_in[0];
    const float* q      = (const float*)d_in[1];
    const float* w      = (const float*)d_in[2];
    const float* b      = (const float*)d_in[3];
    const float* gammas = (const float*)d_in[4];
    const float* G      = (const float*)d_in[5];
    const float* bias   = (const float*)d_in[6];
    const float* qmin   = (const float*)d_in[7];
    float* out  = (float*)d_out;
    float* betw = (float*)d_ws;                 // N*M floats of scratch

    const int n_rows = in_sizes[1] / MODES;     // q is [N, M]

    // Kernel 1: one thread per (row, mode)
    const long long bthreads = (long long)n_rows * MODES;
    const int bblocks = (int)((bthreads + 255) / 256);
    EMMA_beta_kernel<<<bblocks, 256, 0, stream>>>(q, w, b, gammas, G, bias, qmin,
                                                  betw, n_rows);

    // Kernel 2: grid-stride streaming scale (persistent-ish for async pipelining)
    const long long total4 = (long long)n_rows * (MODES * 128 / 4);
    EMMA_scale_kernel<<<8192, 256, 0, stream>>>(x, betw, out, total4);
}